// RegLoss_90194313216574
// MI455X (gfx1250) — compile-verified
//
#include <hip/hip_runtime.h>

// ---------------------------------------------------------------------------
// Masked SmoothL1 + GIoU loss, B*N = 6.4M boxes, D=4.
// Memory-bound streaming reduction: ~230 MB read once -> one f32 scalar.
// Strategy: coalesced b128 loads with NT hints, per-thread accumulation,
// wave-level reduction via V_WMMA_F32_16X16X4_F32 (full fp32), per-block
// partials in d_ws, deterministic single-block finalize.
// ---------------------------------------------------------------------------

typedef __attribute__((ext_vector_type(4))) float vf4;
typedef __attribute__((ext_vector_type(2))) float v2f;
typedef __attribute__((ext_vector_type(8))) float v8f;

#define NBLK 2048
#define NTHR 256
#define NWAVE (NTHR / 32)

// Full 32-lane sum using the fp32 WMMA:
//   A(16x4) = per-lane partial p in both A VGPRs  (each lane holds 2 A elems)
//   B(4x16) = 0.5 everywhere
//   rowsum_m = 0.5 * sum_k A[m][k];  sum over all 16 rows = sum_l p_l.
// D layout: vgpr v holds row v (lanes 0-15) and row v+8 (lanes 16-31), so
// summing d0..d7 gives rows 0-7 in the low half, rows 8-15 in the high half;
// one xor-16 shuffle completes the reduction (result in every lane).
__device__ __forceinline__ float wave_sum32(float p) {
    v2f a; a[0] = p;    a[1] = p;
    v2f b; b[0] = 0.5f; b[1] = 0.5f;
    v8f c = {};
    v8f d = __builtin_amdgcn_wmma_f32_16x16x4_f32(
        /*neg_a=*/false, a, /*neg_b=*/false, b,
        /*c_mod=*/(short)0, c, /*reuse_a=*/false, /*reuse_b=*/false);
    float t = ((d[0] + d[1]) + (d[2] + d[3])) + ((d[4] + d[5]) + (d[6] + d[7]));
    t += __shfl_xor(t, 16, 32);
    return t;
}

__global__ void __launch_bounds__(NTHR)
reg_giou_partial(const vf4* __restrict__ pred,
                 const vf4* __restrict__ tgt,
                 const int* __restrict__ mask,
                 float* __restrict__ ws,
                 int nbox) {
    float sl1 = 0.0f, gsum = 0.0f, cnt = 0.0f;

    const int stride = gridDim.x * blockDim.x;
    for (int i = blockIdx.x * blockDim.x + threadIdx.x; i < nbox; i += stride) {
        vf4 p4 = __builtin_nontemporal_load(pred + i);
        vf4 g4 = __builtin_nontemporal_load(tgt + i);
        int mi = __builtin_nontemporal_load(mask + i);
        float m = mi ? 1.0f : 0.0f;

        // masked boxes (cx, cy, w, h)
        float pcx = p4[0] * m, pcy = p4[1] * m, pw = p4[2] * m, ph = p4[3] * m;
        float gcx = g4[0] * m, gcy = g4[1] * m, gw = g4[2] * m, gh = g4[3] * m;

        // smooth-L1 over the 4 components
        float d0 = pcx - gcx, d1 = pcy - gcy, d2 = pw - gw, d3 = ph - gh;
        float a0 = fabsf(d0), a1 = fabsf(d1), a2 = fabsf(d2), a3 = fabsf(d3);
        sl1 += (a0 < 1.0f) ? 0.5f * d0 * d0 : a0 - 0.5f;
        sl1 += (a1 < 1.0f) ? 0.5f * d1 * d1 : a1 - 0.5f;
        sl1 += (a2 < 1.0f) ? 0.5f * d2 * d2 : a2 - 0.5f;
        sl1 += (a3 < 1.0f) ? 0.5f * d3 * d3 : a3 - 0.5f;

        // cxcywh -> xyxy
        float px0 = pcx - 0.5f * pw, py0 = pcy - 0.5f * ph;
        float px1 = pcx + 0.5f * pw, py1 = pcy + 0.5f * ph;
        float gx0 = gcx - 0.5f * gw, gy0 = gcy - 0.5f * gh;
        float gx1 = gcx + 0.5f * gw, gy1 = gcy + 0.5f * gh;

        // intersection
        float ltx = fmaxf(px0, gx0), lty = fmaxf(py0, gy0);
        float rbx = fminf(px1, gx1), rby = fminf(py1, gy1);
        float iw = fmaxf(rbx - ltx, 0.0f), ih = fmaxf(rby - lty, 0.0f);
        float inter = iw * ih;

        float areap = (px1 - px0) * (py1 - py0);
        float areag = (gx1 - gx0) * (gy1 - gy0);
        float uni = areap + areag - inter;

        // enclosing box
        float cx0 = fminf(px0, gx0), cy0 = fminf(py0, gy0);
        float cx1 = fmaxf(px1, gx1), cy1 = fmaxf(py1, gy1);
        float cw = fmaxf(cx1 - cx0, 0.0f), ch = fmaxf(cy1 - cy0, 0.0f);
        float areac = cw * ch;

        // invalid lanes: substitute 1.0 (matches reference's where-guards)
        float us  = mi ? uni   : 1.0f;
        float acs = mi ? areac : 1.0f;
        float giou = inter / us - (acs - us) / acs;
        gsum += mi ? (1.0f - giou) : 0.0f;
        cnt  += m;
    }

    // wave-level reduction via WMMA (EXEC all-ones here: no early exits)
    float w0 = wave_sum32(sl1);
    float w1 = wave_sum32(gsum);
    float w2 = wave_sum32(cnt);

    __shared__ float sred[3][NWAVE];
    const int lane = threadIdx.x & 31;
    const int wave = threadIdx.x >> 5;
    if (lane == 0) {
        sred[0][wave] = w0;
        sred[1][wave] = w1;
        sred[2][wave] = w2;
    }
    __syncthreads();
    if (threadIdx.x == 0) {
        float b0 = 0.0f, b1 = 0.0f, b2 = 0.0f;
        #pragma unroll
        for (int w = 0; w < NWAVE; ++w) {
            b0 += sred[0][w];
            b1 += sred[1][w];
            b2 += sred[2][w];
        }
        ws[blockIdx.x]            = b0;  // smooth-L1 partial
        ws[NBLK + blockIdx.x]     = b1;  // giou partial
        ws[2 * NBLK + blockIdx.x] = b2;  // valid count partial
    }
}

__global__ void __launch_bounds__(NTHR)
reg_giou_final(const float* __restrict__ ws,
               const int* __restrict__ w_regr,
               const int* __restrict__ w_giou,
               float* __restrict__ out) {
    float s0 = 0.0f, s1 = 0.0f, s2 = 0.0f;
    for (int i = threadIdx.x; i < NBLK; i += NTHR) {
        s0 += ws[i];
        s1 += ws[NBLK + i];
        s2 += ws[2 * NBLK + i];
    }

    float w0 = wave_sum32(s0);
    float w1 = wave_sum32(s1);
    float w2 = wave_sum32(s2);

    __shared__ float sred[3][NWAVE];
    const int lane = threadIdx.x & 31;
    const int wave = threadIdx.x >> 5;
    if (lane == 0) {
        sred[0][wave] = w0;
        sred[1][wave] = w1;
        sred[2][wave] = w2;
    }
    __syncthreads();
    if (threadIdx.x == 0) {
        float t0 = 0.0f, t1 = 0.0f, t2 = 0.0f;
        #pragma unroll
        for (int w = 0; w < NWAVE; ++w) {
            t0 += sred[0][w];
            t1 += sred[1][w];
            t2 += sred[2][w];
        }
        float inv_num = 1.0f / (t2 + 1e-9f);
        float regr_loss = t0 * inv_num;
        float giou_loss = t1 * inv_num;
        out[0] = (float)(*w_regr) * regr_loss + (float)(*w_giou) * giou_loss;
    }
}

extern "C" void kernel_launch(void* const* d_in, const int* in_sizes, int n_in,
                              void* d_out, int out_size, void* d_ws, size_t ws_size,
                              hipStream_t stream) {
    const vf4* pred   = (const vf4*)d_in[0];
    const vf4* tgt    = (const vf4*)d_in[1];
    const int* mask   = (const int*)d_in[2];
    const int* w_regr = (const int*)d_in[3];
    const int* w_giou = (const int*)d_in[4];
    const int  nbox   = in_sizes[2];  // B * N

    float* ws = (float*)d_ws;  // needs 3 * NBLK * 4 = 24 KB

    reg_giou_partial<<<NBLK, NTHR, 0, stream>>>(pred, tgt, mask, ws, nbox);
    reg_giou_final<<<1, NTHR, 0, stream>>>(ws, w_regr, w_giou, (float*)d_out);
}